// RouteOptimizer_27874337751361
// MI455X (gfx1250) — compile-verified
//
#include <hip/hip_runtime.h>

// ---------------------------------------------------------------------------
// RouteOptimizer rollout for MI455X (gfx1250, wave32).
//
//   dist^2(v,d) = [vx, vy, |v|^2, 1] . [-2dx, -2dy, 1, |d|^2]
//
// -> exact rank-4 factorization computed with V_WMMA_F32_16X16X4_F32.
// Each wave owns 16 debris particles (duplicated across both half-waves so
// the B operand is built with cndmasks, no shuffles). Vessel trajectory is
// precomputed per step directly in WMMA A-operand lane layout and staged in
// LDS. Deterministic two-pass reduction (no float atomics) so replayed
// graph launches are bitwise reproducible.
//
// Constant-folding refinements (from asm inspection):
//  * The A operand is pre-scaled by -log2(e)/(2*sigma^2), so the WMMA
//    emits D = scale*dist^2 directly in exp2 domain: v_exp_f32 consumes the
//    WMMA destination registers with no intermediate pk_mul stage.
//  * sin(K*x) computed as v_sin_f32(x * K/(2pi)) = v_sin_f32(x * 0.005)
//    via __builtin_amdgcn_sinf/cosf: one fused constant instead of the
//    K-mul plus the 1/(2pi) normalization __sinf emitted.
//  * 4 round-robin accumulators (compiler packs them into v_pk_add_f32).
// ---------------------------------------------------------------------------

#define T_STEPS 72
#define NV 8
#define BLOCK 256
#define M_PER_BLOCK 128   // 8 waves * 16 debris

typedef __attribute__((ext_vector_type(2))) float v2f;
typedef __attribute__((ext_vector_type(8))) float v8f;

__device__ __constant__ const float kFieldA  = 0.5f;
// FIELD_K = pi/100; v_sin_f32 computes sin(2*pi*t)  =>  turns = x * K/(2pi) = x/200.
__device__ __constant__ const float kTurns   = 0.005f;
// -1/(2*sigma^2) * log2(e), folded into the WMMA A operand.
__device__ __constant__ const float kExp2Scl = -0.02f * 1.4426950408889634f;

// ---------------------------------------------------------------------------
// Kernel 1: sequential vessel rollout (8 vessels, 72 steps) + emit per-step
// WMMA A-operand, pre-scaled by kExp2Scl, in exact 16x4 f32 lane layout:
//   a0: lanes 0-15 -> A[M=lane][K=0] = s*vx ; lanes 16-31 -> A[M][K=2] = s*|v|^2
//   a1: lanes 0-15 -> A[M=lane][K=1] = s*vy ; lanes 16-31 -> A[M][K=3] = s
// Rows 8..15 are zero padding (their D contributions land in lanes 16-31 of
// the C/D layout and are masked out of the reduction).
// ---------------------------------------------------------------------------
__global__ void __launch_bounds__(32)
vessel_precompute(const float* __restrict__ vpos0,
                  const float* __restrict__ controls,
                  float* __restrict__ wsA)
{
    __shared__ float sx[NV], sy[NV];
    int lane = threadIdx.x;                 // 0..31, one wave
    float px = 0.f, py = 0.f;
    if (lane < NV) { px = vpos0[2 * lane]; py = vpos0[2 * lane + 1]; }

    for (int t = 0; t < T_STEPS; ++t) {
        if (lane < NV) {
            float u = -kFieldA * __builtin_amdgcn_sinf(kTurns * px)
                              * __builtin_amdgcn_cosf(kTurns * py);
            float v =  kFieldA * __builtin_amdgcn_cosf(kTurns * px)
                              * __builtin_amdgcn_sinf(kTurns * py);
            float tx = 1.5f * tanhf(controls[(t * NV + lane) * 2 + 0]);
            float ty = 1.5f * tanhf(controls[(t * NV + lane) * 2 + 1]);
            px += u + tx;
            py += v + ty;
            sx[lane] = px;
            sy[lane] = py;
        }
        __syncthreads();
        float a0, a1;
        if (lane < 16) {
            a0 = (lane < NV) ? kExp2Scl * sx[lane] : 0.f;
            a1 = (lane < NV) ? kExp2Scl * sy[lane] : 0.f;
        } else {
            int vv = lane - 16;
            a0 = (vv < NV) ? kExp2Scl * __fmaf_rn(sx[vv], sx[vv], sy[vv] * sy[vv]) : 0.f;
            a1 = (vv < NV) ? kExp2Scl : 0.f;
        }
        wsA[t * 64 + lane]      = a0;
        wsA[t * 64 + 32 + lane] = a1;
        __syncthreads();
    }
}

// ---------------------------------------------------------------------------
// Kernel 2: main rollout. Each wave holds 16 debris (duplicated in both
// half-waves). Per step: advect (v_sin/v_cos), build B operand with two
// cndmasks, one v_wmma_f32_16x16x4_f32 -> 16x16 tile of pre-scaled squared
// distances, 8x v_exp_f32 straight off the WMMA destination into 4
// independent accumulators. Fixed-order LDS block reduction -> partial[blk].
// ---------------------------------------------------------------------------
__global__ void __launch_bounds__(BLOCK)
collection_kernel(const float* __restrict__ dpos0,
                  const float* __restrict__ wsA,
                  float* __restrict__ partial,
                  int M)
{
    __shared__ float sA[T_STEPS * 64];
    __shared__ float red[BLOCK];

    int tid = threadIdx.x;

    // Prefetch + stage all 72 per-step A-operands into LDS once (18 KB).
    __builtin_prefetch(&wsA[tid * (T_STEPS * 64 / BLOCK)], 0, 0);
    for (int i = tid; i < T_STEPS * 64; i += BLOCK) sA[i] = wsA[i];

    int  lane = tid & 31;
    int  col  = lane & 15;                          // debris column in tile
    long wv   = (long)blockIdx.x * (BLOCK / 32) + (tid >> 5);
    long dIdx = wv * 16 + col;
    bool valid = dIdx < (long)M;
    long di    = valid ? dIdx : 0;

    float x = dpos0[2 * di];
    float y = dpos0[2 * di + 1];
    bool  lo = (lane < 16);
    float acc0 = 0.f, acc1 = 0.f, acc2 = 0.f, acc3 = 0.f;

    __syncthreads();

    for (int t = 0; t < T_STEPS; ++t) {
        // Explicit-Euler advection through the gyre field (old position).
        // v_sin_f32(t) = sin(2*pi*t); kTurns folds FIELD_K and 1/(2*pi).
        float tx = kTurns * x, ty = kTurns * y;
        float sx = __builtin_amdgcn_sinf(tx), cx = __builtin_amdgcn_cosf(tx);
        float sy = __builtin_amdgcn_sinf(ty), cy = __builtin_amdgcn_cosf(ty);
        x = __fmaf_rn(-kFieldA, sx * cy, x);
        y = __fmaf_rn( kFieldA, cx * sy, y);

        // A operand (post-update vessel state for this step) from LDS.
        v2f a;
        a.x = sA[t * 64 + lane];
        a.y = sA[t * 64 + 32 + lane];

        // B operand, exact 4x16 f32 layout: lanes 0-15 hold K=0,1; lanes
        // 16-31 hold K=2,3 for the same columns (debris duplicated there).
        float dsq = __fmaf_rn(x, x, y * y);
        v2f b;
        b.x = lo ? (-2.f * x) : 1.f;
        b.y = lo ? (-2.f * y) : dsq;

        v8f c = {};
        // D[v][m] = s*(vx*(-2dx) + vy*(-2dy) + |v|^2 + |d|^2) = s*dist^2,
        // already in exp2 domain (s = -log2e/(2*sigma^2) baked into A).
        v8f d = __builtin_amdgcn_wmma_f32_16x16x4_f32(
            /*neg_a=*/false, a, /*neg_b=*/false, b,
            /*c_mod=*/(short)0, c, /*reuse_a=*/false, /*reuse_b=*/false);

        // v_exp_f32 directly on the WMMA destination registers. Real rows
        // (v=0..7) live entirely in lanes 0-15 of the C/D layout, so these
        // 8 exps waste no instruction slots on padding. Four accumulators
        // keep the adds off the exp dependency chain.
        acc0 += __builtin_amdgcn_exp2f(d[0]);
        acc1 += __builtin_amdgcn_exp2f(d[1]);
        acc2 += __builtin_amdgcn_exp2f(d[2]);
        acc3 += __builtin_amdgcn_exp2f(d[3]);
        acc0 += __builtin_amdgcn_exp2f(d[4]);
        acc1 += __builtin_amdgcn_exp2f(d[5]);
        acc2 += __builtin_amdgcn_exp2f(d[6]);
        acc3 += __builtin_amdgcn_exp2f(d[7]);
    }

    float acc = (acc0 + acc1) + (acc2 + acc3);

    // Lanes 16-31 carry padding rows; invalid columns masked too.
    red[tid] = (lo && valid) ? acc : 0.f;
    __syncthreads();
#pragma unroll
    for (int off = BLOCK / 2; off > 0; off >>= 1) {
        if (tid < off) red[tid] += red[tid + off];
        __syncthreads();
    }
    if (tid == 0) partial[blockIdx.x] = red[0];
}

// ---------------------------------------------------------------------------
// Kernel 3: deterministic final reduction + fuel term.
//   loss = -sum(collection) + 0.001 * sum(controls^2)
// ---------------------------------------------------------------------------
__global__ void __launch_bounds__(BLOCK)
finalize_kernel(const float* __restrict__ partial, int nPart,
                const float* __restrict__ controls, int nCtl,
                float* __restrict__ out)
{
    __shared__ float red[BLOCK];
    int tid = threadIdx.x;
    float s = 0.f;
    for (int i = tid; i < nPart; i += BLOCK) s += partial[i];
    float f = 0.f;
    for (int i = tid; i < nCtl; i += BLOCK) {
        float cc = controls[i];
        f = __fmaf_rn(cc, cc, f);
    }
    red[tid] = -s + 0.001f * f;
    __syncthreads();
#pragma unroll
    for (int off = BLOCK / 2; off > 0; off >>= 1) {
        if (tid < off) red[tid] += red[tid + off];
        __syncthreads();
    }
    if (tid == 0) out[0] = red[0];
}

// ---------------------------------------------------------------------------
extern "C" void kernel_launch(void* const* d_in, const int* in_sizes, int n_in,
                              void* d_out, int out_size, void* d_ws, size_t ws_size,
                              hipStream_t stream)
{
    const float* vpos0    = (const float*)d_in[0];   // (8, 2)
    const float* dpos0    = (const float*)d_in[1];   // (400000, 2)
    const float* controls = (const float*)d_in[2];   // (72, 8, 2)
    float* out = (float*)d_out;
    float* ws  = (float*)d_ws;

    float* wsA     = ws;                    // 72*64 floats (A operands)
    float* partial = ws + T_STEPS * 64;     // per-block partial sums

    int M       = in_sizes[1] / 2;          // 400000
    int nCtl    = in_sizes[2];              // 1152
    int nBlocks = (M + M_PER_BLOCK - 1) / M_PER_BLOCK;   // 3125

    vessel_precompute<<<1, 32, 0, stream>>>(vpos0, controls, wsA);
    collection_kernel<<<nBlocks, BLOCK, 0, stream>>>(dpos0, wsA, partial, M);
    finalize_kernel<<<1, BLOCK, 0, stream>>>(partial, nBlocks, controls, nCtl, out);
}